// MoE_26912265076925
// MI455X (gfx1250) — compile-verified
//
#include <hip/hip_runtime.h>

// Problem constants (fixed by the reference): B=4, S=2048, H=1024, E=8, top-2.
constexpr int H    = 1024;
constexpr int E    = 8;
constexpr int NTOK = 4 * 2048;   // B*S = 8192 tokens

typedef float v2f __attribute__((ext_vector_type(2)));
typedef float v8f __attribute__((ext_vector_type(8)));
typedef int   v4i __attribute__((vector_size(16)));

// ---------------------------------------------------------------------------
// Kernel 1: gating. One wave32 per token.
// ---------------------------------------------------------------------------
__global__ __launch_bounds__(256) void moe_gate_kernel(
    const float* __restrict__ x, const float* __restrict__ gate_w,
    const float* __restrict__ gate_b, const float* __restrict__ expert_b,
    float* __restrict__ out, int* __restrict__ counts,
    int* __restrict__ tokList, float* __restrict__ gateList) {
  const int lane  = threadIdx.x & 31;
  const int token = blockIdx.x * 8 + (threadIdx.x >> 5);
  if (token >= NTOK) return;

  float acc[E];
#pragma unroll
  for (int e = 0; e < E; ++e) acc[e] = 0.f;

  const float4* x4  = (const float4*)(x + (size_t)token * H);
  const float4* gw4 = (const float4*)gate_w;
#pragma unroll
  for (int i = 0; i < H / 128; ++i) {
    const int idx = i * 32 + lane;
    const float4 xv = x4[idx];
#pragma unroll
    for (int e = 0; e < E; ++e) {
      const float4 g = gw4[e * (H / 4) + idx];
      acc[e] += xv.x * g.x + xv.y * g.y + xv.z * g.z + xv.w * g.w;
    }
  }
  // wave32 butterfly: every lane ends with the full dot product.
#pragma unroll
  for (int off = 16; off >= 1; off >>= 1)
#pragma unroll
    for (int e = 0; e < E; ++e) acc[e] += __shfl_xor(acc[e], off, 32);

  float mx = acc[0] + gate_b[0];
  float logit[E];
#pragma unroll
  for (int e = 0; e < E; ++e) {
    logit[e] = acc[e] + gate_b[e];
    mx = fmaxf(mx, logit[e]);
  }
  float sum = 0.f, g[E];
#pragma unroll
  for (int e = 0; e < E; ++e) { g[e] = __expf(logit[e] - mx); sum += g[e]; }
  const float inv = 1.f / sum;
#pragma unroll
  for (int e = 0; e < E; ++e) g[e] *= inv;

  // top-2 (strict > : first occurrence wins ties, matching jax top_k)
  int i1 = 0; float v1 = g[0];
#pragma unroll
  for (int e = 1; e < E; ++e) if (g[e] > v1) { v1 = g[e]; i1 = e; }
  int i2 = (i1 == 0) ? 1 : 0; float v2 = g[i2];
#pragma unroll
  for (int e = 0; e < E; ++e)
    if (e != i1 && g[e] > v2) { v2 = g[e]; i2 = e; }

  if (lane == 0) {
    int p1 = atomicAdd(&counts[i1], 1);
    tokList[i1 * NTOK + p1]  = token;
    gateList[i1 * NTOK + p1] = v1;
    int p2 = atomicAdd(&counts[i2], 1);
    tokList[i2 * NTOK + p2]  = token;
    gateList[i2 * NTOK + p2] = v2;
  }

  // out[t] = g1*b[e1] + g2*b[e2]  (also overwrites harness poison)
  float4* o4 = (float4*)(out + (size_t)token * H);
  const float4* b1 = (const float4*)(expert_b + (size_t)i1 * H);
  const float4* b2 = (const float4*)(expert_b + (size_t)i2 * H);
#pragma unroll
  for (int i = 0; i < H / 128; ++i) {
    const int idx = i * 32 + lane;
    const float4 a = b1[idx], b = b2[idx];
    float4 r;
    r.x = v1 * a.x + v2 * b.x;  r.y = v1 * a.y + v2 * b.y;
    r.z = v1 * a.z + v2 * b.z;  r.w = v1 * a.w + v2 * b.w;
    o4[idx] = r;
  }
}

// ---------------------------------------------------------------------------
// Kernel 2: expert-grouped GEMM, fp32 WMMA, async global->LDS double buffering
// grid = (H/128 col tiles, NTOK/32 row tiles, E); block = 256 (8 waves)
// Block tile: 32 gathered tokens x 128 cols, K chunks of 64.
// Each wave: 16x32 tile = two v8f accumulators of v_wmma_f32_16x16x4_f32.
// ---------------------------------------------------------------------------
constexpr int LDS_STRIDE = 68;          // 64 + 4 pad: conflict-free, 16B align
constexpr int A_SZ   = 32 * LDS_STRIDE;
constexpr int B_SZ   = 128 * LDS_STRIDE;
constexpr int NCHUNK = H / 64;          // 16 K-chunks
// per-wave async instructions per chunk: 2 (A) + 8 (B) = 10

__device__ __forceinline__ void async_copy16(const float* g, float* l) {
  __builtin_amdgcn_global_load_async_to_lds_b128(
      (__attribute__((address_space(1))) v4i*)(g),
      (__attribute__((address_space(3))) v4i*)(l),
      /*offset=*/0, /*cpol=*/0);
}

__global__ __launch_bounds__(256) void moe_expert_gemm_kernel(
    const float* __restrict__ x, const float* __restrict__ expert_w,
    float* __restrict__ out, const int* __restrict__ counts,
    const int* __restrict__ tokList, const float* __restrict__ gateList) {
  const int e = blockIdx.z;
  const int rows = counts[e];
  const int tileStart = blockIdx.y * 32;
  if (tileStart >= rows) return;                  // inactive tile: early exit

  const int tid  = threadIdx.x;
  const int lane = tid & 31;
  const int wv   = tid >> 5;
  const int ln   = lane & 15;                     // M/N within 16-wide tile
  const int lh   = lane >> 4;                     // selects K pair (ISA 7.12.2)

  __shared__ float As[2 * A_SZ];                  // [buf][32 rows x 64 K]
  __shared__ float Bs[2 * B_SZ];                  // [buf][128 N x 64 K] (W^T)

  const int colBlock = blockIdx.x * 128;

  // --- per-thread fill assignments (resolved once) ---------------------------
  // A: 512 float4 per chunk -> 2 per thread; rows r0 = tid>>4, r1 = r0+16.
  const int aq = (tid & 15) * 4;                  // K offset within chunk
  const int r0 = tid >> 4, r1 = r0 + 16;
  const int p0 = tileStart + r0, p1 = tileStart + r1;
  const int t0 = tokList[e * NTOK + (p0 < rows ? p0 : tileStart)];
  const int t1 = tokList[e * NTOK + (p1 < rows ? p1 : tileStart)];
  const float* gA0 = x + (size_t)t0 * H + aq;
  const float* gA1 = x + (size_t)t1 * H + aq;
  float* lA0 = &As[r0 * LDS_STRIDE + aq];
  float* lA1 = &As[r1 * LDS_STRIDE + aq];
  // B: 2048 float4 per chunk -> 8 per thread; cols n = (tid>>4) + it*16.
  const float* gB = expert_w + ((size_t)e * H + colBlock + r0) * H + aq;
  float* lB = &Bs[r0 * LDS_STRIDE + aq];

  // --- async fill of one K-chunk into buffer `buf` ---------------------------
  auto issueFills = [&](int chunk) {
    const int buf = chunk & 1;
    const int k0  = chunk * 64;
    async_copy16(gA0 + k0, lA0 + buf * A_SZ);
    async_copy16(gA1 + k0, lA1 + buf * A_SZ);
#pragma unroll
    for (int it = 0; it < 8; ++it)
      async_copy16(gB + (size_t)(it * 16) * H + k0,
                   lB + it * 16 * LDS_STRIDE + buf * B_SZ);
  };

  const int waveRow = 16 * (wv & 1);              // 0 or 16
  const int waveCol = 32 * (wv >> 1);             // 0,32,64,96
  v8f acc0 = {};
  v8f acc1 = {};

  issueFills(0);
  for (int c = 0; c < NCHUNK; ++c) {
    const int buf = c & 1;
    if (c + 1 < NCHUNK) {
      issueFills(c + 1);                          // prefetch next chunk
      __builtin_amdgcn_s_wait_asynccnt(10);       // current chunk landed
    } else {
      __builtin_amdgcn_s_wait_asynccnt(0);
    }
    __syncthreads();

    const float* Ab = As + buf * A_SZ;
    const float* Bb = Bs + buf * B_SZ;
#pragma unroll
    for (int kk = 0; kk < 64; kk += 4) {
      const int ka = kk + 2 * lh;  // lanes 0-15: K{kk,kk+1}; 16-31: K{kk+2,+3}
      const v2f a  = *(const v2f*)&Ab[(waveRow + ln) * LDS_STRIDE + ka];
      const v2f b0 = *(const v2f*)&Bb[(waveCol + ln) * LDS_STRIDE + ka];
      const v2f b1 = *(const v2f*)&Bb[(waveCol + 16 + ln) * LDS_STRIDE + ka];
      acc0 = __builtin_amdgcn_wmma_f32_16x16x4_f32(
          false, a, false, b0, (short)0, acc0, false, false);
      acc1 = __builtin_amdgcn_wmma_f32_16x16x4_f32(
          false, a, false, b1, (short)0, acc1, false, false);
    }
    __syncthreads();
  }

  // C/D layout (ISA 7.12.2): VGPR r -> M = r + 8*lh, N = ln.
  const int col0 = colBlock + waveCol + ln;
#pragma unroll
  for (int r = 0; r < 8; ++r) {
    const int mm  = waveRow + r + 8 * lh;
    const int pos = tileStart + mm;
    if (pos < rows) {
      const int   t = tokList[e * NTOK + pos];
      const float g = gateList[e * NTOK + pos];
      atomicAdd(out + (size_t)t * H + col0,      g * acc0[r]);
      atomicAdd(out + (size_t)t * H + col0 + 16, g * acc1[r]);
    }
  }
}

// ---------------------------------------------------------------------------
extern "C" void kernel_launch(void* const* d_in, const int* in_sizes, int n_in,
                              void* d_out, int out_size, void* d_ws,
                              size_t ws_size, hipStream_t stream) {
  const float* x        = (const float*)d_in[0];
  const float* gate_w   = (const float*)d_in[1];
  const float* gate_b   = (const float*)d_in[2];
  const float* expert_w = (const float*)d_in[3];
  const float* expert_b = (const float*)d_in[4];
  float* out = (float*)d_out;

  int*   counts   = (int*)d_ws;                                    // 8 ints
  int*   tokList  = (int*)((char*)d_ws + 1024);                    // E*NTOK
  float* gateList = (float*)((char*)d_ws + 1024 +
                             sizeof(int) * (size_t)E * NTOK);      // E*NTOK

  (void)hipMemsetAsync(counts, 0, 1024, stream);

  moe_gate_kernel<<<NTOK / 8, 256, 0, stream>>>(
      x, gate_w, gate_b, expert_b, out, counts, tokList, gateList);

  dim3 grid(H / 128, NTOK / 32, E);
  moe_expert_gemm_kernel<<<grid, 256, 0, stream>>>(
      x, expert_w, out, counts, tokList, gateList);
}